// TGS_2439541424656
// MI455X (gfx1250) — compile-verified
//
#include <hip/hip_runtime.h>

// CDNA5 / gfx1250 fused 3-level tree-GNN propagation.
// h[l+1][i] = relu( [h[l][i], h[l][i]+data_last[nbr[l][i]]] @ W^T + b )
// Fused across levels (h[i] only depends on h[i] + constants) using
// V_WMMA_F32_16X16X4_F32 for exact-fp32 matrix math.

typedef __attribute__((ext_vector_type(2))) float v2f;
typedef __attribute__((ext_vector_type(8))) float v8f;

#define DFEAT   64
#define KTOT    128
#define XSTRIDE 132          // 128 cols + 4 pad floats: kills LDS bank conflicts
#define WAVES   8            // 256-thread blocks

__global__ __launch_bounds__(256) void tgs_fused_kernel(
    const float* __restrict__ x0,
    const float* __restrict__ dlast,
    const long long* __restrict__ nbr,
    const float* __restrict__ W,
    const float* __restrict__ bias,
    float* __restrict__ out,
    int N, int L, int numTiles)
{
    // B fragments of W^T in wmma order: [kk(32)][ntile(4)][lane(32)][2]  = 32 KB
    __shared__ float sBf[32 * 4 * 32 * 2];
    // per-wave A staging slab: 16 rows x (128+4) floats                  = 67.6 KB
    __shared__ __align__(16) float sX[WAVES][16 * XSTRIDE];
    __shared__ float sBias[DFEAT];
    __shared__ long long sIdx[WAVES][16];

    const int tid = threadIdx.x;

    // ---- stage W^T into WMMA B-fragment layout ----
    // fragment (kk, nt, lane): b.x = Wt[kk*4 + 2*(lane>>4)][nt*16 + (lane&15)]
    //                          b.y = same k + 1           (Wt[k][n] = W[n*128+k])
    for (int idx = tid; idx < 32 * 4 * 32; idx += 256) {
        int l  = idx & 31;
        int nt = (idx >> 5) & 3;
        int kk = idx >> 7;
        int n  = nt * 16 + (l & 15);
        int k  = kk * 4 + 2 * (l >> 4);
        sBf[idx * 2 + 0] = W[n * KTOT + k];
        sBf[idx * 2 + 1] = W[n * KTOT + k + 1];
    }
    if (tid < DFEAT) sBias[tid] = bias[tid];
    __syncthreads();

    const int wave = tid >> 5;
    const int lane = tid & 31;
    const int g    = lane >> 4;   // half-wave group (K-split / M-split)
    const int nl   = lane & 15;
    float* xw = sX[wave];

    const int waveGlobal = blockIdx.x * WAVES + wave;
    const int waveStride = gridDim.x * WAVES;

    for (int tile = waveGlobal; tile < numTiles; tile += waveStride) {
        const int rowBase = tile * 16;

        // ---- load 16x64 h tile (level-0 input) into cols 0..63 ----
        for (int i = lane; i < 256; i += 32) {
            int m  = i >> 4;
            int c4 = (i & 15) * 4;
            int nr = rowBase + m; if (nr >= N) nr = N - 1;
            float4 hv = *(const float4*)(x0 + (size_t)nr * DFEAT + c4);
            *(float4*)(xw + m * XSTRIDE + c4) = hv;
        }

        for (int l = 0; l < L; ++l) {
            // ---- stage neighbor indices for this tile/level ----
            if (lane < 16) {
                int nr = rowBase + lane; if (nr >= N) nr = N - 1;
                sIdx[wave][lane] = nbr[(size_t)l * N + nr];
            }
            // ---- gather + h_tilde = h + data_last[j] into cols 64..127 ----
            for (int i = lane; i < 256; i += 32) {
                int m  = i >> 4;
                int c4 = (i & 15) * 4;
                long long j = sIdx[wave][m];
                float4 gv = *(const float4*)(dlast + (size_t)j * DFEAT + c4);
                float4 hv = *(const float4*)(xw + m * XSTRIDE + c4);
                hv.x += gv.x; hv.y += gv.y; hv.z += gv.z; hv.w += gv.w;
                *(float4*)(xw + m * XSTRIDE + DFEAT + c4) = hv;
            }

            // ---- 16x64 = (16x128)(128x64) via v_wmma_f32_16x16x4_f32 ----
            v8f acc[4];
            acc[0] = (v8f){}; acc[1] = (v8f){}; acc[2] = (v8f){}; acc[3] = (v8f){};
            // A fragment: lane(0..15)=row M, VGPR0=K+0,VGPR1=K+1; lanes 16..31: K+2,K+3
            const float* arow = xw + nl * XSTRIDE + 2 * g;
            #pragma unroll 8
            for (int kk = 0; kk < 32; ++kk) {
                v2f a;
                a.x = arow[kk * 4 + 0];
                a.y = arow[kk * 4 + 1];
                const float* bf = sBf + ((kk * 4) * 32 + lane) * 2;  // [kk][0][lane]
                v2f b0; b0.x = bf[0];   b0.y = bf[1];
                v2f b1; b1.x = bf[64];  b1.y = bf[65];
                v2f b2; b2.x = bf[128]; b2.y = bf[129];
                v2f b3; b3.x = bf[192]; b3.y = bf[193];
                acc[0] = __builtin_amdgcn_wmma_f32_16x16x4_f32(false, a, false, b0, (short)0, acc[0], false, false);
                acc[1] = __builtin_amdgcn_wmma_f32_16x16x4_f32(false, a, false, b1, (short)0, acc[1], false, false);
                acc[2] = __builtin_amdgcn_wmma_f32_16x16x4_f32(false, a, false, b2, (short)0, acc[2], false, false);
                acc[3] = __builtin_amdgcn_wmma_f32_16x16x4_f32(false, a, false, b3, (short)0, acc[3], false, false);
            }

            // ---- bias + relu; write back (LDS for inner levels, global last) ----
            const bool last = (l == L - 1);
            #pragma unroll
            for (int nt = 0; nt < 4; ++nt) {
                float bb = sBias[nt * 16 + nl];
                #pragma unroll
                for (int r = 0; r < 8; ++r) {
                    // C/D layout: VGPR r -> M = r + 8*g, N = nt*16 + nl
                    float v = fmaxf(acc[nt][r] + bb, 0.0f);
                    int m = r + 8 * g;
                    if (last) {
                        int nr = rowBase + m;
                        if (nr < N)
                            out[(size_t)nr * DFEAT + nt * 16 + nl] = v;
                    } else {
                        xw[m * XSTRIDE + nt * 16 + nl] = v;
                    }
                }
            }
            // same-wave LDS ops are in-order: next level's reads see these writes
        }
    }
}

extern "C" void kernel_launch(void* const* d_in, const int* in_sizes, int n_in,
                              void* d_out, int out_size, void* d_ws, size_t ws_size,
                              hipStream_t stream) {
    const float*      x0    = (const float*)d_in[0];
    const float*      dlast = (const float*)d_in[1];
    const long long*  nbr   = (const long long*)d_in[2];   // int64 in reference
    const float*      W     = (const float*)d_in[3];       // [64, 128] row-major
    const float*      bias  = (const float*)d_in[4];
    float*            out   = (float*)d_out;

    const int N = in_sizes[0] / DFEAT;
    const int L = (N > 0) ? (in_sizes[2] / N) : 0;
    const int numTiles = (N + 15) / 16;

    int blocks = 1280;                      // persistent-ish: 10240 waves
    if (blocks * WAVES > numTiles) blocks = (numTiles + WAVES - 1) / WAVES;
    if (blocks < 1) blocks = 1;

    tgs_fused_kernel<<<blocks, 256, 0, stream>>>(x0, dlast, nbr, W, bias, out,
                                                 N, L, numTiles);
}